// kmeans_68564857913657
// MI455X (gfx1250) — compile-verified
//
#include <hip/hip_runtime.h>
#include <hip/hip_bf16.h>

#define NC 128
#define KDIM 64
#define PTS_PER_BLOCK 128   // 8 waves * 16 points
#define THREADS 256

typedef float v2f __attribute__((ext_vector_type(2)));
typedef float v8f __attribute__((ext_vector_type(8)));

// ---------------------------------------------------------------------------
// zero a float buffer
// ---------------------------------------------------------------------------
__global__ void kz_zero(float* __restrict__ p, int n) {
    int i = blockIdx.x * blockDim.x + threadIdx.x;
    if (i < n) p[i] = 0.0f;
}

// ---------------------------------------------------------------------------
// mu[c][k] = sums[c][k] / counts[c]
// ---------------------------------------------------------------------------
__global__ void kz_update(const float* __restrict__ sums,
                          const float* __restrict__ counts,
                          float* __restrict__ mu) {
    int i = blockIdx.x * blockDim.x + threadIdx.x;
    if (i < NC * KDIM) {
        mu[i] = sums[i] / counts[i >> 6];
    }
}

// ---------------------------------------------------------------------------
// Assignment (+ optional scatter-mean accumulation) pass.
//   dot = mu @ X^T via V_WMMA_F32_16X16X4_F32  (A = 16x4 mu tile, B = 4x16 X^T)
//   label[p] = argmin_c (mu2[c] - 2*dot[c][p])      (x2[p] is argmin-invariant)
// ---------------------------------------------------------------------------
template <bool ACCUM, bool WRITE_LABELS>
__global__ __launch_bounds__(THREADS)
void kz_assign(const float* __restrict__ X,
               const float* __restrict__ mu,
               float* __restrict__ sums,     // (NC,K) global accumulators
               float* __restrict__ counts,   // (NC,)
               float* __restrict__ labels,   // (N,) written as float
               int n) {
    __shared__ float smem[NC * KDIM];   // mu during assign; partial sums during accum
    __shared__ float mu2_s[NC];
    __shared__ float cnt_s[NC];
    __shared__ int   label_s[PTS_PER_BLOCK];

    const int t = threadIdx.x;

    // ---- stage mu into LDS (32 KB) ----
    {
        const float4* src = reinterpret_cast<const float4*>(mu);
        float4* dst = reinterpret_cast<float4*>(smem);
        #pragma unroll
        for (int i = 0; i < (NC * KDIM / 4) / THREADS; ++i)
            dst[t + i * THREADS] = src[t + i * THREADS];
    }
    __syncthreads();

    if (t < NC) {
        float s = 0.0f;
        #pragma unroll
        for (int k = 0; k < KDIM; ++k) {
            float v = smem[t * KDIM + k];
            s += v * v;
        }
        mu2_s[t] = s;
    }
    __syncthreads();

    const int wave = t >> 5;
    const int lane = t & 31;
    const int pbase = blockIdx.x * PTS_PER_BLOCK + wave * 16;
    const int myp   = pbase + (lane & 15);
    const int pclmp = myp < n ? myp : (n - 1);      // clamp: keep EXEC all-ones for WMMA
    const int koff  = (lane >> 4) * 2;              // lanes 0-15: k+0,k+1 ; lanes 16-31: k+2,k+3

    // ---- preload B fragments: X^T 4x16 slices, one float2 per lane per k-step ----
    v2f bfrag[KDIM / 4];
    {
        const float* xrow = X + (size_t)pclmp * KDIM + koff;
        #pragma unroll
        for (int ks = 0; ks < KDIM / 4; ++ks)
            bfrag[ks] = *reinterpret_cast<const v2f*>(xrow + ks * 4);
    }

    // ---- 8 cluster tiles x 16 k-steps of v_wmma_f32_16x16x4_f32 ----
    float bestv = 3.402823466e38f;
    int   bestc = 0;
    const int chi = (lane >> 4) * 8;   // D rows: M = r + 8*(lane>=16)
    #pragma unroll
    for (int ct = 0; ct < NC / 16; ++ct) {
        const float* amu = smem + (ct * 16 + (lane & 15)) * KDIM + koff;
        v8f acc = {};
        #pragma unroll
        for (int ks = 0; ks < KDIM / 4; ++ks) {
            v2f a = *reinterpret_cast<const v2f*>(amu + ks * 4);
            acc = __builtin_amdgcn_wmma_f32_16x16x4_f32(
                false, a, false, bfrag[ks], (short)0, acc, false, false);
        }
        #pragma unroll
        for (int r = 0; r < 8; ++r) {
            int   c = ct * 16 + chi + r;
            float v = mu2_s[c] - 2.0f * acc[r];
            if (v < bestv) { bestv = v; bestc = c; }
        }
    }

    // ---- merge the two half-wave partials for each point (wave32 shfl) ----
    {
        float ov = __shfl_xor(bestv, 16);
        int   oc = __shfl_xor(bestc, 16);
        if (ov < bestv || (ov == bestv && oc < bestc)) { bestv = ov; bestc = oc; }
    }

    const int lp = wave * 16 + (lane & 15);
    if (lane < 16) label_s[lp] = (myp < n) ? bestc : -1;
    if (WRITE_LABELS && lane < 16 && myp < n) labels[myp] = (float)bestc;

    if (ACCUM) {
        __syncthreads();
        // reuse smem as per-block partial sums
        #pragma unroll
        for (int i = 0; i < (NC * KDIM) / THREADS; ++i)
            smem[t + i * THREADS] = 0.0f;
        if (t < NC) cnt_s[t] = 0.0f;
        __syncthreads();

        // cooperative scatter-add: each thread handles 32 (point,k) pairs
        #pragma unroll 4
        for (int i = 0; i < (PTS_PER_BLOCK * KDIM) / THREADS; ++i) {
            int flat = t + i * THREADS;
            int lpt  = flat >> 6;       // local point
            int k    = flat & 63;
            int c    = label_s[lpt];
            if (c >= 0) {
                float xv = X[(size_t)(blockIdx.x * PTS_PER_BLOCK + lpt) * KDIM + k];
                unsafeAtomicAdd(&smem[c * KDIM + k], xv);   // ds_add_f32
            }
        }
        if (t < PTS_PER_BLOCK) {
            int c = label_s[t];
            if (c >= 0) unsafeAtomicAdd(&cnt_s[c], 1.0f);
        }
        __syncthreads();

        // flush block partials to global accumulators
        #pragma unroll 4
        for (int i = 0; i < (NC * KDIM) / THREADS; ++i) {
            int flat = t + i * THREADS;
            unsafeAtomicAdd(&sums[flat], smem[flat]);       // global_atomic_add_f32
        }
        if (t < NC) unsafeAtomicAdd(&counts[t], cnt_s[t]);
    }
}

// ---------------------------------------------------------------------------
// host-side orchestration
// ---------------------------------------------------------------------------
extern "C" void kernel_launch(void* const* d_in, const int* in_sizes, int n_in,
                              void* d_out, int out_size, void* d_ws, size_t ws_size,
                              hipStream_t stream) {
    const float* X   = (const float*)d_in[0];
    const float* mu0 = (const float*)d_in[1];
    const int n = in_sizes[0] / KDIM;           // 500000
    const int niter = 5;                        // fixed by setup_inputs

    float* mu_cur = (float*)d_ws;               // NC*K
    float* sums   = mu_cur + NC * KDIM;         // NC*K
    float* counts = sums + NC * KDIM;           // NC   (contiguous after sums)

    float* out_mu  = (float*)d_out;
    float* out_lab = out_mu + NC * KDIM;

    hipMemcpyAsync(mu_cur, mu0, NC * KDIM * sizeof(float),
                   hipMemcpyDeviceToDevice, stream);

    const int ablocks = (n + PTS_PER_BLOCK - 1) / PTS_PER_BLOCK;
    const int zcount  = NC * KDIM + NC;

    for (int it = 0; it < niter; ++it) {
        kz_zero<<<(zcount + THREADS - 1) / THREADS, THREADS, 0, stream>>>(sums, zcount);
        kz_assign<true, false><<<ablocks, THREADS, 0, stream>>>(
            X, mu_cur, sums, counts, nullptr, n);
        kz_update<<<(NC * KDIM + THREADS - 1) / THREADS, THREADS, 0, stream>>>(
            sums, counts, mu_cur);
    }
    // final labeling pass
    kz_assign<false, true><<<ablocks, THREADS, 0, stream>>>(
        X, mu_cur, sums, counts, out_lab, n);

    hipMemcpyAsync(out_mu, mu_cur, NC * KDIM * sizeof(float),
                   hipMemcpyDeviceToDevice, stream);
}